// LTCCell_82884278878574
// MI455X (gfx1250) — compile-verified
//
#include <hip/hip_runtime.h>
#include <cstdint>

// LTC cell, B=512, N=256, S=128, M=64, 6 ODE unfolds.
// Trans-pipe bound: 201M gate evals -> 402M v_exp_f32/v_rcp_f32. Packed
// constants (1.5MB) are L2-resident; HBM traffic ~3MB. 128 blocks x 1024
// threads, 4-way i-split for 8 waves/SIMD of exp/rcp latency hiding.
// The gate itself is not a contraction (depends on b,i,n jointly), but the
// cross-group partial combine  ns[n,bb] = sum_q p[q][n,bb]  is an exact
// unit-weight f32 contraction -> mapped to V_WMMA_F32_16X16X4_F32 (K=4=NQ).

#define B_  512
#define N_  256
#define S_  128
#define M_  64
#define UNF 6
#define BT  4             // batch tile per block -> 128 blocks
#define NQ  4             // i-range split (thread groups per block)
#define TPB (N_ * NQ)     // 1024 threads = 32 wave32s
#define LOG2E 1.4426950408889634f

#define P_ELEMS (N_ * N_)
#define Q_ELEMS (S_ * N_)
#define WS_NEEDED ((size_t)(P_ELEMS + Q_ELEMS) * sizeof(float4))

typedef float v2f __attribute__((ext_vector_type(2)));
typedef float v8f __attribute__((ext_vector_type(8)));

// ---- pre-pass: fold (w*mask), (w*mask*E), (sig*log2e), (sig*log2e*mu) ------
__global__ __launch_bounds__(256)
void pack_kernel(const float* __restrict__ w,    const float* __restrict__ sig,
                 const float* __restrict__ mu,   const float* __restrict__ erev,
                 const float* __restrict__ mask,
                 const float* __restrict__ sw,   const float* __restrict__ ssig,
                 const float* __restrict__ smu,  const float* __restrict__ serev,
                 const float* __restrict__ smask,
                 float4* __restrict__ P, float4* __restrict__ Q)
{
    const int idx = blockIdx.x * 256 + threadIdx.x;
    if (idx < P_ELEMS) {
        const float m = w[idx] * mask[idx];
        const float a = sig[idx] * LOG2E;
        P[idx] = make_float4(m, m * erev[idx], a, a * mu[idx]);
    }
    if (idx < Q_ELEMS) {
        const float m = sw[idx] * smask[idx];
        const float a = ssig[idx] * LOG2E;
        Q[idx] = make_float4(m, m * serev[idx], a, a * smu[idx]);
    }
}

template <bool PACKED>
__global__ __launch_bounds__(TPB, 1)
void ltc_kernel(const float* __restrict__ x,      // (B,S)
                const float* __restrict__ state,  // (B,N)
                const float* __restrict__ gleak,  // (N)
                const float* __restrict__ eleak,  // (N)
                const float* __restrict__ cm,     // (N)
                const float* __restrict__ w,      // (N,N) [i*N+n]
                const float* __restrict__ sig,
                const float* __restrict__ mu,
                const float* __restrict__ erev,
                const float* __restrict__ sw,     // (S,N)
                const float* __restrict__ ssig,
                const float* __restrict__ smu,
                const float* __restrict__ serev,
                const float* __restrict__ mask,
                const float* __restrict__ smask,
                const float* __restrict__ iw,     // (S)
                const float* __restrict__ ib,     // (S)
                const float* __restrict__ ow,     // (M)
                const float* __restrict__ ob,     // (M)
                const float4* __restrict__ P,     // packed recurrent consts
                const float4* __restrict__ Q,     // packed sensory consts
                float* __restrict__ out)          // (B*M) ++ (B*N)
{
    __shared__ __align__(16) float  sv [N_][BT];  // voltages, transposed
    __shared__ __align__(16) float  sxt[S_][BT];  // mapped inputs, transposed
    __shared__ __align__(16) float4 pnum[NQ][N_]; // partial numerators
    __shared__ __align__(16) float4 pden[NQ][N_]; // partial denominators
    __shared__ float scmt[N_], sglel[N_], sdbase[N_]; // per-neuron ODE consts

    const int tid = threadIdx.x;
    const int n   = tid & (N_ - 1);               // neuron column
    const int q   = tid >> 8;                     // i/s-range quarter
    const int b0  = blockIdx.x * BT;              // batch tile base

    // ---- Stage state -> LDS via gfx1250 async-to-LDS path ------------------
    {
        const uint32_t sv_base = (uint32_t)(uintptr_t)(&sv[0][0]);
        for (int k = tid; k < BT * N_; k += TPB) {
            const int bb = k >> 8;
            const int i  = k & (N_ - 1);
            uint32_t lds_addr = sv_base + 4u * (uint32_t)(i * BT + bb);
            const float* gp = state + (size_t)(b0 + bb) * N_ + i;
            asm volatile("global_load_async_to_lds_b32 %0, %1, off"
                         :: "v"(lds_addr), "v"(gp) : "memory");
        }
    }
    // ---- xt = x*iw + ib ----------------------------------------------------
    for (int k = tid; k < BT * S_; k += TPB) {
        const int bb = k >> 7;
        const int s  = k & (S_ - 1);
        sxt[s][bb] = fmaf(x[(size_t)(b0 + bb) * S_ + s], iw[s], ib[s]);
    }
    // ---- per-neuron ODE constants into LDS (used by WMMA-combine lanes) ----
    if (tid < N_) {
        const float cmt = cm[tid] * (float)UNF;   // cm / (elapsed/unfolds)
        const float gl  = gleak[tid];
        scmt[tid]   = cmt;
        sglel[tid]  = gl * eleak[tid];
        sdbase[tid] = cmt + gl + 1e-8f;           // EPS folded in
    }
    asm volatile("s_wait_asynccnt 0x0" ::: "memory");
    __syncthreads();

    // ---- Sensory aggregation: this group's s-quarter (constant over unfolds)
    float wns[BT], wds[BT];
    #pragma unroll
    for (int bb = 0; bb < BT; ++bb) { wns[bb] = 0.f; wds[bb] = 0.f; }

    const int slo = q * (S_ / NQ), shi = slo + (S_ / NQ);
    #pragma unroll 2
    for (int s = slo; s < shi; ++s) {
        const int idx = s * N_ + n;
        float m, mE, a, c;
        if (PACKED) {
            const float4 p = Q[idx];
            m = p.x; mE = p.y; a = p.z; c = p.w;
        } else {
            m  = sw[idx] * smask[idx];
            mE = m * serev[idx];
            a  = ssig[idx] * LOG2E;
            c  = a * smu[idx];
        }
        const float4 xv = *(const float4*)&sxt[s][0];
        const float xs[BT] = {xv.x, xv.y, xv.z, xv.w};
        #pragma unroll
        for (int bb = 0; bb < BT; ++bb) {
            // sigmoid(sig*(x-mu)) = 1 / (1 + 2^(c - a*x))
            const float e = __builtin_amdgcn_exp2f(fmaf(-a, xs[bb], c));
            const float g = __builtin_amdgcn_rcpf(1.0f + e);
            wds[bb] = fmaf(m,  g, wds[bb]);
            wns[bb] = fmaf(mE, g, wns[bb]);
        }
    }

    const int ilo = q * (N_ / NQ), ihi = ilo + (N_ / NQ);
    const int lane = tid & 31;
    const int wv   = tid >> 5;                    // wave id (0..31)
    const int qsel = (lane >> 4) << 1;            // B-matrix K striping: 0 or 2
    const float* pnf = (const float*)&pnum[0][0]; // flat [q*1024 + n*4 + bb]
    const float* pdf = (const float*)&pden[0][0];

    // ---- ODE unfolds -------------------------------------------------------
    for (int u = 0; u < UNF; ++u) {
        float num[BT], den[BT];
        #pragma unroll
        for (int bb = 0; bb < BT; ++bb) { num[bb] = wns[bb]; den[bb] = wds[bb]; }

        #pragma unroll 2
        for (int i = ilo; i < ihi; ++i) {
            const int idx = i * N_ + n;
            if ((i & 31) == 0) {                   // keep L2 lines in flight
                const int pi = i + 32;
                if (pi < ihi) {
                    const int pidx = pi * N_ + n;
                    if (PACKED) {
                        __builtin_prefetch(&P[pidx], 0, 3);
                    } else {
                        __builtin_prefetch(&w[pidx],    0, 3);
                        __builtin_prefetch(&sig[pidx],  0, 3);
                        __builtin_prefetch(&mu[pidx],   0, 3);
                        __builtin_prefetch(&erev[pidx], 0, 3);
                        __builtin_prefetch(&mask[pidx], 0, 3);
                    }
                }
            }
            float m, mE, a, c;
            if (PACKED) {
                const float4 p = P[idx];           // one global_load_b128
                m = p.x; mE = p.y; a = p.z; c = p.w;
            } else {
                m  = w[idx] * mask[idx];
                mE = m * erev[idx];
                a  = sig[idx] * LOG2E;
                c  = a * mu[idx];
            }
            const float4 vv = *(const float4*)&sv[i][0];   // ds_load_b128 bcast
            const float vs[BT] = {vv.x, vv.y, vv.z, vv.w};
            #pragma unroll
            for (int bb = 0; bb < BT; ++bb) {
                const float e = __builtin_amdgcn_exp2f(fmaf(-a, vs[bb], c));
                const float g = __builtin_amdgcn_rcpf(1.0f + e);
                den[bb] = fmaf(m,  g, den[bb]);
                num[bb] = fmaf(mE, g, num[bb]);
            }
        }

        pnum[q][n] = make_float4(num[0], num[1], num[2], num[3]);
        pden[q][n] = make_float4(den[0], den[1], den[2], den[3]);
        __syncthreads();   // partials visible; all sv reads for unfold u done

        // ---- exact unit-weight contraction over q on the matrix pipe -------
        // D(16x16) = Ones(16x4) x B(4x16); column j of D = sum_q p[q][j].
        // Wave wv covers flattened outputs j = n*4+bb in [wv*32, wv*32+32).
        {
            const v2f Aones = {1.0f, 1.0f};
            const v8f Cz = {};
            #pragma unroll
            for (int p2 = 0; p2 < 2; ++p2) {
                const int jl = wv * 32 + p2 * 16 + (lane & 15);
                const v2f Bn = { pnf[qsel * (N_ * BT) + jl],
                                 pnf[(qsel + 1) * (N_ * BT) + jl] };
                const v2f Bd = { pdf[qsel * (N_ * BT) + jl],
                                 pdf[(qsel + 1) * (N_ * BT) + jl] };
                const v8f Dn = __builtin_amdgcn_wmma_f32_16x16x4_f32(
                    false, Aones, false, Bn, (short)0, Cz, false, false);
                const v8f Dd = __builtin_amdgcn_wmma_f32_16x16x4_f32(
                    false, Aones, false, Bd, (short)0, Cz, false, false);
                if (lane < 16) {                  // lanes 16-31 hold duplicates
                    const int n2 = jl >> 2, bb2 = jl & 3;
                    const float vold = sv[n2][bb2];
                    sv[n2][bb2] = (fmaf(scmt[n2], vold, sglel[n2]) + Dn[0]) /
                                  (sdbase[n2] + Dd[0]);
                }
            }
        }
        __syncthreads();   // v updated + partials consumed before next unfold
    }

    // ---- Outputs: tuple (out(B,M), v(B,N)) concatenated --------------------
    if (tid < N_) {
        #pragma unroll
        for (int bb = 0; bb < BT; ++bb)
            out[(size_t)B_ * M_ + (size_t)(b0 + bb) * N_ + n] = sv[n][bb];
        if (n < M_) {
            const float owv = ow[n], obv = ob[n];
            #pragma unroll
            for (int bb = 0; bb < BT; ++bb)
                out[(size_t)(b0 + bb) * M_ + n] = fmaf(sv[n][bb], owv, obv);
        }
    }
}

extern "C" void kernel_launch(void* const* d_in, const int* in_sizes, int n_in,
                              void* d_out, int out_size, void* d_ws, size_t ws_size,
                              hipStream_t stream) {
    (void)in_sizes; (void)n_in; (void)out_size;
    const float* x      = (const float*)d_in[0];
    const float* state  = (const float*)d_in[1];
    const float* gleak  = (const float*)d_in[2];
    const float* eleak  = (const float*)d_in[3];
    const float* cm     = (const float*)d_in[4];
    const float* w      = (const float*)d_in[5];
    const float* sig    = (const float*)d_in[6];
    const float* mu     = (const float*)d_in[7];
    const float* erev   = (const float*)d_in[8];
    const float* sw     = (const float*)d_in[9];
    const float* ssig   = (const float*)d_in[10];
    const float* smu    = (const float*)d_in[11];
    const float* serev  = (const float*)d_in[12];
    const float* mask   = (const float*)d_in[13];
    const float* smask  = (const float*)d_in[14];
    const float* iw     = (const float*)d_in[15];
    const float* ib     = (const float*)d_in[16];
    const float* ow     = (const float*)d_in[17];
    const float* ob     = (const float*)d_in[18];
    float* out          = (float*)d_out;

    if (ws_size >= WS_NEEDED) {
        float4* Pp = (float4*)d_ws;
        float4* Qp = Pp + P_ELEMS;
        pack_kernel<<<dim3((P_ELEMS + 255) / 256), dim3(256), 0, stream>>>(
            w, sig, mu, erev, mask, sw, ssig, smu, serev, smask, Pp, Qp);
        ltc_kernel<true><<<dim3(B_ / BT), dim3(TPB), 0, stream>>>(
            x, state, gleak, eleak, cm, w, sig, mu, erev,
            sw, ssig, smu, serev, mask, smask, iw, ib, ow, ob, Pp, Qp, out);
    } else {
        ltc_kernel<false><<<dim3(B_ / BT), dim3(TPB), 0, stream>>>(
            x, state, gleak, eleak, cm, w, sig, mu, erev,
            sw, ssig, smu, serev, mask, smask, iw, ib, ow, ob,
            (const float4*)nullptr, (const float4*)nullptr, out);
    }
}